// SAGENET_10943576670342
// MI455X (gfx1250) — compile-verified
//
#include <hip/hip_runtime.h>

// ---------------- problem constants (match reference) ----------------
#define N_NODES   50000          // == 3125 * 16, M dimension tiles exactly
#define N_EDGES   800000
#define N_FEAT    100
#define DPAD      112            // 100 padded to 7*16 for WMMA tiling
#define N_GRAPHS  256
#define N_CLASSES 2

typedef __attribute__((ext_vector_type(2))) float v2f;   // A/B frag of 16x16x4 f32 WMMA
typedef __attribute__((ext_vector_type(8))) float v8f;   // C/D frag

typedef unsigned int u32x4 __attribute__((ext_vector_type(4)));
typedef int          i32x4 __attribute__((ext_vector_type(4)));
typedef int          i32x8 __attribute__((ext_vector_type(8)));

static inline int cdiv(int a, int b) { return (a + b - 1) / b; }

// ---------------- utility kernels ----------------
__global__ void fill_zero_kernel(float* __restrict__ p, int n) {
  int i = blockIdx.x * blockDim.x + threadIdx.x;
  if (i < n) p[i] = 0.0f;
}

__global__ void pad_features_kernel(const float* __restrict__ x, float* __restrict__ A) {
  int i = blockIdx.x * blockDim.x + threadIdx.x;     // over N_NODES*DPAD
  if (i >= N_NODES * DPAD) return;
  int n = i / DPAD, f = i - n * DPAD;
  A[i] = (f < N_FEAT) ? x[n * N_FEAT + f] : 0.0f;
}

__global__ void pad_weight_kernel(const float* __restrict__ W, float* __restrict__ Wp) {
  int i = blockIdx.x * blockDim.x + threadIdx.x;     // over DPAD*DPAD
  if (i >= DPAD * DPAD) return;
  int k = i / DPAD, n = i - k * DPAD;
  Wp[i] = (k < N_FEAT && n < N_FEAT) ? W[k * N_FEAT + n] : 0.0f;
}

__global__ void pad_bias_kernel(const float* __restrict__ b, float* __restrict__ bp) {
  int i = blockIdx.x * blockDim.x + threadIdx.x;
  if (i < DPAD) bp[i] = (i < N_FEAT) ? b[i] : 0.0f;
}

// ---------------- graph structure (computed once) ----------------
__global__ void degree_kernel(const int* __restrict__ dst, float* __restrict__ cnt) {
  int e = blockIdx.x * blockDim.x + threadIdx.x;
  if (e < N_EDGES) atomicAdd(&cnt[dst[e]], 1.0f);
}

__global__ void pool_count_kernel(const int* __restrict__ batch, float* __restrict__ gc) {
  int n = blockIdx.x * blockDim.x + threadIdx.x;
  if (n < N_NODES) atomicAdd(&gc[batch[n]], 1.0f);
}

// ---------------- edge gather + scatter-add (L2-resident atomics) ----------------
// one thread per (edge, 4-feature quad); 25 quads cover features 0..99
__global__ void scatter_kernel(const int* __restrict__ src, const int* __restrict__ dst,
                               const float* __restrict__ A, float* __restrict__ S) {
  int t = blockIdx.x * blockDim.x + threadIdx.x;     // N_EDGES * 25
  if (t >= N_EDGES * 25) return;
  int e = t / 25, q = t - e * 25;
  int s = src[e], d = dst[e];
  const float4 v = *(const float4*)&A[s * DPAD + q * 4];
  float* o = &S[d * DPAD + q * 4];
  atomicAdd(o + 0, v.x);
  atomicAdd(o + 1, v.y);
  atomicAdd(o + 2, v.z);
  atomicAdd(o + 3, v.w);
}

// ---------------- TDM: stage one 112x112 f32 matrix from global into LDS ----------------
// Builds a 2D Tensor DMA Descriptor (D#):
//   group0: count=1 | lds_addr | 57-bit global_addr | type=2
//   group1: data_size=4B, tensor_dim0=tensor_dim1=112, tile_dim0=tile_dim1=112,
//           tensor_dim0_stride=112 (elements); groups 2/3 zero (<=2D tensor)
__device__ __forceinline__ void tdm_load_matrix(const float* gptr, void* lds_ptr) {
#if defined(__has_builtin) && __has_builtin(__builtin_amdgcn_tensor_load_to_lds)
  unsigned long long ga = (unsigned long long)gptr;
  unsigned lds_off = (unsigned)(unsigned long long)lds_ptr;   // flat LDS aperture: [31:0] = LDS byte offset

  u32x4 g0;
  g0[0] = 1u;                                         // count=1, is_restore=0, gather_mode=0
  g0[1] = lds_off;                                    // lds_addr (bytes)
  g0[2] = (unsigned)(ga & 0xffffffffu);               // global_addr[31:0]
  g0[3] = (unsigned)((ga >> 32) & 0x01ffffffu)        // global_addr[56:32]
        | (2u << 30);                                 // type=2 ("image")

  i32x8 g1;
  g1[0] = (int)(2u << 16);                            // workgroup_mask=0, data_size=2 (4 bytes)
  g1[1] = (int)((unsigned)DPAD << 16);                // tensor_dim0[15:0] @bits 63:48
  g1[2] = (int)((unsigned)DPAD << 16);                // tensor_dim1[15:0] @bits 95:80
  g1[3] = (int)((unsigned)DPAD << 16);                // tile_dim0 @bits 127:112
  g1[4] = (int)DPAD;                                  // tile_dim1 @bits 143:128 (tile_dim2=0)
  g1[5] = (int)DPAD;                                  // tensor_dim0_stride[31:0] (elements)
  g1[6] = 0;                                          // stride[47:32]=0, dim1_stride lo = 0
  g1[7] = 0;

  i32x4 gz = {0, 0, 0, 0};
#if __clang_major__ >= 23
  i32x8 gz8 = {0, 0, 0, 0, 0, 0, 0, 0};
  __builtin_amdgcn_tensor_load_to_lds(g0, g1, gz, gz, gz8, 0);
#else
  __builtin_amdgcn_tensor_load_to_lds(g0, g1, gz, gz, 0);
#endif
#else
  (void)gptr; (void)lds_ptr;
#endif
}

// ---------------- fused SAGE GEMM: H = act( (S*inv_deg)@Wl + A@Wr + b ) ----------------
// Weights TDM-staged into LDS once per workgroup; one wave32 per 16x16 output tile;
// K = DPAD in steps of 4 via V_WMMA_F32_16X16X4_F32 (A from VMEM, B from LDS).
__global__ __launch_bounds__(256)
void sage_gemm_kernel(const float* __restrict__ S, const float* __restrict__ cnt,
                      const float* __restrict__ A,
                      const float* __restrict__ Wl, const float* __restrict__ Wr,
                      const float* __restrict__ bias, float* __restrict__ H, int relu) {
  __shared__ float sWl[DPAD * DPAD];   // 50,176 B
  __shared__ float sWr[DPAD * DPAD];   // 50,176 B  (100 KB total; 320 KB/WGP budget)

  const int NT_M = N_NODES / 16;   // 3125
  const int NT_N = DPAD / 16;      // 7

#if defined(__has_builtin) && __has_builtin(__builtin_amdgcn_tensor_load_to_lds)
  if (threadIdx.x == 0) {                       // TDM ignores EXEC; issue once per workgroup
    tdm_load_matrix(Wl, sWl);
    tdm_load_matrix(Wr, sWr);
    __builtin_amdgcn_s_wait_tensorcnt(0);       // issuing wave owns TENSORcnt
  }
#else
  for (int i = threadIdx.x; i < DPAD * DPAD; i += blockDim.x) {   // cooperative fallback
    sWl[i] = Wl[i];
    sWr[i] = Wr[i];
  }
#endif
  __syncthreads();                              // all waves see staged weights

  int wave = (blockIdx.x * blockDim.x + threadIdx.x) >> 5;   // wave-uniform
  int lane = threadIdx.x & 31;
  if (wave >= NT_M * NT_N) return;              // wave-uniform exit: EXEC all-1s below
  int tm = wave / NT_N, tn = wave - tm * NT_N;
  int m0 = tm * 16, n0 = tn * 16;

  // A-frag (16x4 f32): lanes 0-15 hold K={0,1}, lanes 16-31 hold K={2,3}; row = m0 + (lane&15)
  // B-frag (4x16 f32): same K split; col = n0 + (lane&15)
  int row   = m0 + (lane & 15);
  int col   = n0 + (lane & 15);
  int khalf = (lane < 16) ? 0 : 2;
  float inv = 1.0f / fmaxf(cnt[row], 1.0f);     // mean-aggregation scale fused into A load

  v8f c = {};
  for (int k = 0; k < DPAD; k += 4) {
    int kk = k + khalf;
    v2f aS = { S[row * DPAD + kk] * inv, S[row * DPAD + kk + 1] * inv };
    v2f aX = { A[row * DPAD + kk],       A[row * DPAD + kk + 1] };
    v2f bL = { sWl[kk * DPAD + col],     sWl[(kk + 1) * DPAD + col] };   // ds_load
    v2f bR = { sWr[kk * DPAD + col],     sWr[(kk + 1) * DPAD + col] };   // ds_load
    // (neg_a, A, neg_b, B, c_mod, C, reuse_a, reuse_b)
    c = __builtin_amdgcn_wmma_f32_16x16x4_f32(false, aS, false, bL, (short)0, c, false, false);
    c = __builtin_amdgcn_wmma_f32_16x16x4_f32(false, aX, false, bR, (short)0, c, false, false);
  }

  // C/D layout: VGPR r -> M = r + (lane<16 ? 0 : 8), N = lane&15
  float bcol = bias[col];
#pragma unroll
  for (int r = 0; r < 8; ++r) {
    int orow = m0 + r + ((lane < 16) ? 0 : 8);
    float v = c[r] + bcol;
    if (relu) v = fmaxf(v, 0.0f);
    H[orow * DPAD + col] = v;
  }
}

// ---------------- global mean pool (sum part) ----------------
__global__ void pool_scatter_kernel(const int* __restrict__ batch, const float* __restrict__ Hf,
                                    float* __restrict__ G) {
  int t = blockIdx.x * blockDim.x + threadIdx.x;   // N_NODES * 25
  if (t >= N_NODES * 25) return;
  int n = t / 25, q = t - n * 25;
  int g = batch[n];
  const float4 v = *(const float4*)&Hf[n * DPAD + q * 4];
  float* o = &G[g * DPAD + q * 4];
  atomicAdd(o + 0, v.x);
  atomicAdd(o + 1, v.y);
  atomicAdd(o + 2, v.z);
  atomicAdd(o + 3, v.w);
}

// ---------------- final linear head: out = (G/gc) @ Wlin + blin ----------------
__global__ void final_linear_kernel(const float* __restrict__ G, const float* __restrict__ gc,
                                    const float* __restrict__ Wlin, const float* __restrict__ blin,
                                    float* __restrict__ out) {
  int t = blockIdx.x * blockDim.x + threadIdx.x;   // N_GRAPHS * N_CLASSES
  if (t >= N_GRAPHS * N_CLASSES) return;
  int g = t / N_CLASSES, cc = t - g * N_CLASSES;
  float inv = 1.0f / fmaxf(gc[g], 1.0f);
  float acc = blin[cc];
  for (int k = 0; k < N_FEAT; ++k)
    acc += G[g * DPAD + k] * inv * Wlin[k * N_CLASSES + cc];
  out[t] = acc;
}

// ---------------- orchestration ----------------
extern "C" void kernel_launch(void* const* d_in, const int* in_sizes, int n_in,
                              void* d_out, int out_size, void* d_ws, size_t ws_size,
                              hipStream_t stream) {
  (void)in_sizes; (void)n_in; (void)out_size; (void)ws_size;

  const float* x    = (const float*)d_in[0];
  const int*   src  = (const int*)d_in[1];            // edge_index row 0 (JAX x64-off => int32)
  const int*   dst  = src + N_EDGES;                  // edge_index row 1
  const int*   batch= (const int*)d_in[2];
  const float* Wl_in[3] = { (const float*)d_in[3], (const float*)d_in[6], (const float*)d_in[9]  };
  const float* b_in [3] = { (const float*)d_in[4], (const float*)d_in[7], (const float*)d_in[10] };
  const float* Wr_in[3] = { (const float*)d_in[5], (const float*)d_in[8], (const float*)d_in[11] };
  const float* Wlin = (const float*)d_in[12];
  const float* blin = (const float*)d_in[13];
  float* out = (float*)d_out;

  // workspace layout (floats)
  float* ws  = (float*)d_ws;
  float* A   = ws;                           // 50000*112
  float* S   = A   + N_NODES * DPAD;         // 50000*112
  float* Hb  = S   + N_NODES * DPAD;         // 50000*112
  float* CNT = Hb  + N_NODES * DPAD;         // 50000
  float* WL  = CNT + N_NODES;                // 112*112
  float* WR  = WL  + DPAD * DPAD;            // 112*112
  float* BB  = WR  + DPAD * DPAD;            // 112
  float* G   = BB  + DPAD;                   // 256*112
  float* GC  = G   + N_GRAPHS * DPAD;        // 256

  const int T = 256;
  const int tiles = (N_NODES / 16) * (DPAD / 16);        // 21875 waves
  const int gemm_blocks = cdiv(tiles, T / 32);

  // pad input features; graph-structure counts (degree + pool) computed once
  pad_features_kernel<<<cdiv(N_NODES * DPAD, T), T, 0, stream>>>(x, A);
  fill_zero_kernel<<<cdiv(N_NODES, T), T, 0, stream>>>(CNT, N_NODES);
  degree_kernel<<<cdiv(N_EDGES, T), T, 0, stream>>>(dst, CNT);
  fill_zero_kernel<<<cdiv(N_GRAPHS, T), T, 0, stream>>>(GC, N_GRAPHS);
  pool_count_kernel<<<cdiv(N_NODES, T), T, 0, stream>>>(batch, GC);

  float* cur = A;
  float* nxt = Hb;
  for (int L = 0; L < 3; ++L) {
    fill_zero_kernel<<<cdiv(N_NODES * DPAD, T), T, 0, stream>>>(S, N_NODES * DPAD);
    scatter_kernel<<<cdiv(N_EDGES * 25, T), T, 0, stream>>>(src, dst, cur, S);
    pad_weight_kernel<<<cdiv(DPAD * DPAD, T), T, 0, stream>>>(Wl_in[L], WL);
    pad_weight_kernel<<<cdiv(DPAD * DPAD, T), T, 0, stream>>>(Wr_in[L], WR);
    pad_bias_kernel<<<1, T, 0, stream>>>(b_in[L], BB);
    sage_gemm_kernel<<<gemm_blocks, T, 0, stream>>>(S, CNT, cur, WL, WR, BB, nxt,
                                                    (L < 2) ? 1 : 0);
    float* tmp = cur; cur = nxt; nxt = tmp;   // 'cur' now holds layer output
  }

  fill_zero_kernel<<<cdiv(N_GRAPHS * DPAD, T), T, 0, stream>>>(G, N_GRAPHS * DPAD);
  pool_scatter_kernel<<<cdiv(N_NODES * 25, T), T, 0, stream>>>(batch, cur, G);
  final_linear_kernel<<<cdiv(N_GRAPHS * N_CLASSES, T), T, 0, stream>>>(G, GC, Wlin, blin, out);
}